// PPO_ac_CSTR1_LearnableBoundsMPC_84739704750206
// MI455X (gfx1250) — compile-verified
//
#include <hip/hip_runtime.h>
#include <hip/hip_bf16.h>

// ---------------------------------------------------------------------------
// Problem constants (from the reference)
// ---------------------------------------------------------------------------
// NZ=8 T_H=36 NX=2 NU=2 NCC=4
// NUOFF=288 NSOFF=358 NXSOFF=394 NSSOFF=466 NVAR=502 -> pad 512
// MCON=769 -> pad 784
#define NVARP 512
#define MCONP 784
#define BATCH 1024
#define N_ITERS 150
#define SIGMA_C 1e-6f
#define INF_C 1e20f

typedef __attribute__((ext_vector_type(2))) float v2f;
typedef __attribute__((ext_vector_type(8))) float v8f;

// ---------------------------------------------------------------------------
// CDNA5 async global->LDS staging (ASYNCcnt-tracked), with safe fallback.
// Builtin signatures (from clang diagnostics):
//   b32 : (AS1 int*        gaddr, AS3 int*        lds, Ii offset, Ii cpol)
//   b128: (AS1 vint4*      gaddr, AS3 vint4*      lds, Ii offset, Ii cpol)
//   where vint4 = int __attribute__((vector_size(16)))
// ---------------------------------------------------------------------------
#if __has_builtin(__builtin_amdgcn_global_load_async_to_lds_b32)
#define HAS_ASYNC32 1
#else
#define HAS_ASYNC32 0
#endif
#if __has_builtin(__builtin_amdgcn_global_load_async_to_lds_b128)
#define HAS_ASYNC128 1
#else
#define HAS_ASYNC128 0
#endif

#if HAS_ASYNC32 || HAS_ASYNC128
typedef int vint4 __attribute__((vector_size(16)));
typedef __attribute__((address_space(1))) int as1_int;
typedef __attribute__((address_space(3))) int as3_int;
typedef __attribute__((address_space(1))) vint4 as1_int4;
typedef __attribute__((address_space(3))) vint4 as3_int4;
#endif

__device__ __forceinline__ void async_copy_b32(const float* g, float* l) {
#if HAS_ASYNC32
  __builtin_amdgcn_global_load_async_to_lds_b32((as1_int*)g, (as3_int*)l, 0, 0);
#else
  *l = *g;
#endif
}

__device__ __forceinline__ void async_copy_b128(const float* g, float* l) {
#if HAS_ASYNC128
  __builtin_amdgcn_global_load_async_to_lds_b128((as1_int4*)g, (as3_int4*)l, 0, 0);
#else
  *(float4*)l = *(const float4*)g;
#endif
}

__device__ __forceinline__ void async_wait_all() {
#if __has_builtin(__builtin_amdgcn_s_wait_asynccnt)
  __builtin_amdgcn_s_wait_asynccnt(0);
#elif HAS_ASYNC32 || HAS_ASYNC128
  asm volatile("s_wait_asynccnt 0x0" ::: "memory");
#endif
}

// workspace layout (float offsets)
constexpr size_t OFF_A    = 0;                                  // 784*512
constexpr size_t OFF_AT   = OFF_A    + (size_t)MCONP * NVARP;   // 512*784
constexpr size_t OFF_MT   = OFF_AT   + (size_t)NVARP * MCONP;   // 512*512
constexpr size_t OFF_MI   = OFF_MT   + (size_t)NVARP * NVARP;   // 512*512
constexpr size_t OFF_RHO  = OFF_MI   + (size_t)NVARP * NVARP;   // 784
constexpr size_t OFF_IRHO = OFF_RHO  + MCONP;                   // 784
constexpr size_t OFF_Q    = OFF_IRHO + MCONP;                   // B*512
constexpr size_t OFF_X    = OFF_Q    + (size_t)BATCH * NVARP;
constexpr size_t OFF_RHS  = OFF_X    + (size_t)BATCH * NVARP;
constexpr size_t OFF_Z    = OFF_RHS  + (size_t)BATCH * NVARP;   // B*784
constexpr size_t OFF_Y    = OFF_Z    + (size_t)BATCH * MCONP;
constexpr size_t OFF_L    = OFF_Y    + (size_t)BATCH * MCONP;
constexpr size_t OFF_U    = OFF_L    + (size_t)BATCH * MCONP;
constexpr size_t TOTAL_F  = OFF_U    + (size_t)BATCH * MCONP;

// ---------------------------------------------------------------------------
__global__ void zero_kernel(float* p, size_t n) {
  size_t i = (size_t)blockIdx.x * blockDim.x + threadIdx.x;
  size_t s = (size_t)gridDim.x * blockDim.x;
  for (; i < n; i += s) p[i] = 0.f;
}

// ---------------------------------------------------------------------------
// Build dense A (784x512, zero-padded, pre-zeroed) + rho / inv_rho.
// ---------------------------------------------------------------------------
__global__ void build_static(const float* __restrict__ Az,
                             const float* __restrict__ Au,
                             const float* __restrict__ ZtoX,
                             float* __restrict__ A,
                             float* __restrict__ rho,
                             float* __restrict__ irho) {
  const int tid = threadIdx.x;
  for (int m = tid; m < MCONP; m += 256) {
    float r = ((m < 9) || (m >= 223 && m < 590)) ? 1000.f : 1.f;
    rho[m] = r;
    irho[m] = 1.f / r;
  }
  for (int j = tid; j < 8; j += 256) A[j * NVARP + j] = 1.f;
  if (tid == 0) A[8 * NVARP + 358] = 1.f;
  for (int idx = tid; idx < 36 * 2 * 8; idx += 256) {
    int t = idx >> 4, k = (idx >> 3) & 1, j = idx & 7;
    float v = ZtoX[k * 8 + j];
    int zr = 2 * t + k, zc = 8 * t + j;
    A[(9 + zr) * NVARP + zc] = v;
    A[(81 + zr) * NVARP + zc] = v;
  }
  for (int i = tid; i < 72; i += 256) {
    A[(9 + i) * NVARP + 394 + i] = 1.f;
    A[(81 + i) * NVARP + 394 + i] = -1.f;
  }
  for (int i = tid; i < 70; i += 256) A[(153 + i) * NVARP + 288 + i] = 1.f;
  for (int idx = tid; idx < 52; idx += 256) {
    int i = idx >> 1, kk = idx & 1;
    int t = 4 * (i / 3) + (i % 3) + 1;
    A[(223 + 2 * i + kk) * NVARP + 288 + 2 * t + kk] = 1.f;
    A[(223 + 2 * i + kk) * NVARP + 288 + 2 * (t - 1) + kk] = -1.f;
  }
  for (int idx = tid; idx < 35 * 8 * 8; idx += 256) {
    int t1 = idx / 64 + 1, j = (idx >> 3) & 7, jz = idx & 7;
    A[(275 + 8 * (t1 - 1) + j) * NVARP + 8 * (t1 - 1) + jz] = Az[j * 8 + jz];
  }
  for (int idx = tid; idx < 35 * 8 * 2; idx += 256) {
    int t1 = idx / 16 + 1, j = (idx >> 1) & 7, ju = idx & 1;
    A[(275 + 8 * (t1 - 1) + j) * NVARP + 288 + 2 * (t1 - 1) + ju] = Au[j * 2 + ju];
  }
  for (int idx = tid; idx < 35 * 8; idx += 256) {
    int t1 = (idx >> 3) + 1, j = idx & 7;
    A[(275 + 8 * (t1 - 1) + j) * NVARP + 8 * t1 + j] = -1.f;
  }
  for (int idx = tid; idx < 35; idx += 256) {
    int t1 = idx + 1, r = 555 + idx;
    A[r * NVARP + 358 + t1] = 1.f;
    A[r * NVARP + 358 + t1 - 1] = -1.f;
    A[r * NVARP + 288 + 2 * idx] = -0.05f;   // -0.2/NCC
  }
  for (int idx = tid; idx < 35; idx += 256) {
    int t1 = idx + 1;
    A[(590 + idx) * NVARP + 358 + t1] = 1.f;
    A[(590 + idx) * NVARP + 466 + t1] = 1.f;
    A[(625 + idx) * NVARP + 358 + t1] = 1.f;
    A[(625 + idx) * NVARP + 466 + t1] = -1.f;
  }
  if (tid == 1) {
    A[660 * NVARP + 358 + 35] = 1.f;
    A[660 * NVARP + 466 + 35] = 1.f;
  }
  for (int i = tid; i < 72; i += 256) A[(661 + i) * NVARP + 394 + i] = 1.f;
  for (int t = tid; t < 36; t += 256) A[(733 + t) * NVARP + 466 + t] = 1.f;
}

// ---------------------------------------------------------------------------
// Per-batch bounds l/u (784 each) and linear cost q (512). One block per row.
// ---------------------------------------------------------------------------
__global__ __launch_bounds__(128) void build_bounds_q(
    const float* __restrict__ feat, const float* __restrict__ Xlb_p,
    const float* __restrict__ Xub_p, const float* __restrict__ slb_p,
    const float* __restrict__ sub_p, const float* __restrict__ tgt_p,
    const float* __restrict__ XtoZ, float* __restrict__ l,
    float* __restrict__ u, float* __restrict__ q) {
  const int b = blockIdx.x;
  __shared__ float zi[8];
  __shared__ float sc[4];
  __shared__ float xlb[2], xub[2];
  if (threadIdx.x == 0) {
    float s0 = feat[b * 12 + 0], s1 = feat[b * 12 + 1];
    for (int j = 0; j < 8; ++j) zi[j] = XtoZ[j * 2] * s0 + XtoZ[j * 2 + 1] * s1;
    sc[0] = feat[b * 12 + 2];
    sc[1] = tanhf(slb_p[0]) * 0.5f;
    sc[2] = tanhf(sub_p[0]) * 0.5f;
    sc[3] = tanhf(tgt_p[0]);
  }
  if (threadIdx.x == 1) {
    xlb[0] = tanhf(Xlb_p[0]) * 0.5f; xlb[1] = tanhf(Xlb_p[1]) * 0.5f;
    xub[0] = tanhf(Xub_p[0]) * 0.5f; xub[1] = tanhf(Xub_p[1]) * 0.5f;
  }
  __syncthreads();
  float* lr = &l[(size_t)b * MCONP];
  float* ur = &u[(size_t)b * MCONP];
  for (int m = threadIdx.x; m < MCONP; m += 128) {
    float lo = -INF_C, hi = INF_C;
    if (m < 8)          { lo = hi = zi[m]; }
    else if (m == 8)    { lo = hi = sc[0]; }
    else if (m < 81)    { lo = -1.f + xlb[(m - 9) & 1]; }
    else if (m < 153)   { hi = 1.f + xub[(m - 81) & 1]; }
    else if (m < 223)   { lo = -1.f; hi = 1.f; }
    else if (m < 590)   { lo = 0.f; hi = 0.f; }
    else if (m < 625)   { lo = sc[1]; }
    else if (m < 660)   { hi = 6.f + sc[2]; }
    else if (m == 660)  { lo = 1.f + sc[3]; }
    else if (m < 769)   { lo = 0.f; }
    else                { lo = 0.f; hi = 0.f; }   // pad rows pinned to 0
    lr[m] = lo; ur[m] = hi;
  }
  float* qr = &q[(size_t)b * NVARP];
  for (int c = threadIdx.x; c < NVARP; c += 128) {
    float v = 0.f;
    if (c >= 289 && c <= 357 && (((c - 289) & 1) == 0)) {
      int t = (c - 289) >> 1;
      v = feat[b * 12 + 3 + (t >> 2)];
    }
    qr[c] = v;
  }
}

// ---------------------------------------------------------------------------
// Mmat = diag(PDIAG + sigma) + A^T diag(rho) A   (padded region = identity)
// ---------------------------------------------------------------------------
__global__ void build_mmat(const float* __restrict__ A,
                           const float* __restrict__ rho,
                           float* __restrict__ M) {
  int idx = blockIdx.x * blockDim.x + threadIdx.x;
  if (idx >= NVARP * NVARP) return;
  int i = idx >> 9, j = idx & 511;
  float s = 0.f;
  for (int m = 0; m < MCONP; ++m)
    s += A[m * NVARP + i] * rho[m] * A[m * NVARP + j];
  if (i == j) {
    if (i >= 502) s += 1.0f;
    else s += (i >= 394 ? 20000.0f : 0.0f) + SIGMA_C;
  }
  M[idx] = s;
}

__global__ void transposeA(const float* __restrict__ A, float* __restrict__ At) {
  int idx = blockIdx.x * blockDim.x + threadIdx.x;
  if (idx >= MCONP * NVARP) return;
  int m = idx / NVARP, k = idx % NVARP;
  At[(size_t)k * MCONP + m] = A[idx];
}

// ---------------------------------------------------------------------------
// Gauss-Jordan inverse of SPD 512x512 (one workgroup, one-time cost).
// ---------------------------------------------------------------------------
__global__ __launch_bounds__(1024) void gauss_jordan(float* __restrict__ M,
                                                     float* __restrict__ Inv) {
  const int tid = threadIdx.x;
  for (int i = tid; i < NVARP * NVARP; i += 1024) Inv[i] = 0.f;
  __syncthreads();
  for (int i = tid; i < NVARP; i += 1024) Inv[i * NVARP + i] = 1.f;
  __syncthreads();
  __shared__ float pm[NVARP], pi[NVARP];
  __shared__ float pinv_s;
  for (int p = 0; p < NVARP; ++p) {
    if (tid == 0) pinv_s = 1.0f / M[p * NVARP + p];
    __syncthreads();
    float pinv = pinv_s;
    for (int c = tid; c < NVARP; c += 1024) {
      float mv = M[p * NVARP + c] * pinv;
      float iv = Inv[p * NVARP + c] * pinv;
      M[p * NVARP + c] = mv; Inv[p * NVARP + c] = iv;
      pm[c] = mv; pi[c] = iv;
    }
    __syncthreads();
    int r = tid >> 1;
    int c0 = (tid & 1) * 256;
    if (r != p) {
      float f = M[r * NVARP + p];
      if (f != 0.f) {
        for (int c = c0; c < c0 + 256; ++c) {
          M[r * NVARP + c]   -= f * pm[c];
          Inv[r * NVARP + c] -= f * pi[c];
        }
      }
    }
    __syncthreads();
  }
}

// ---------------------------------------------------------------------------
// WMMA f32 GEMM:  Out[b,n] = sum_k Xop[b,k] * W[k,n]  (+ fused epilogue)
//   MODE 0: Xop = rho*z - y ; epilogue Out(rhs) = acc + sigma*x - q
//   MODE 1: Xop = rhs       ; Out = x
//   MODE 2: Xop = x, W = A^T; epilogue z = clip(acc + y/rho, l, u),
//                             y += rho*(acc - z)
// Block: 256 thr = 8 wave32. Block tile 128(batch) x 64(N); wave = 16x64.
// Double-buffered LDS; W tile (and X for modes 1/2) staged with CDNA5
// async global->LDS copies (ASYNCcnt), one barrier per k-block.
// ---------------------------------------------------------------------------
template <int MODE>
__global__ __launch_bounds__(256) void admm_gemm(
    const float* __restrict__ Xmat, const float* __restrict__ Wmat,
    float* __restrict__ Outp, float* __restrict__ ybuf,
    const float* __restrict__ rho, const float* __restrict__ irho,
    const float* __restrict__ qbuf, const float* __restrict__ xprev,
    const float* __restrict__ lbuf, const float* __restrict__ ubuf,
    int K, int N, int ldx, int ldw) {
  __shared__ float Xs[2][128 * 20];  // [row][k] stride 20 (bank-safe)
  __shared__ float Ws[2][64 * 20];   // [n][k]  stride 20

  const int tid = threadIdx.x;
  const int wave = tid >> 5;
  const int lane = tid & 31;
  const int bbase = blockIdx.y * 128;
  const int n0 = blockIdx.x * 64;

  v8f acc0 = {0.f,0.f,0.f,0.f,0.f,0.f,0.f,0.f};
  v8f acc1 = acc0, acc2 = acc0, acc3 = acc0;

  const int xr = tid >> 1;         // 0..127 : batch row in tile
  const int xk = (tid & 1) * 8;    // 0 or 8 : k-offset
  const int wk = tid >> 4;         // 0..15  : k for W tile
  const int wn = (tid & 15) * 4;   // 0..60  : n for W tile

  auto stage = [&](int s, int kb) {
    // ---- X tile (128 x 16), k-contiguous in LDS ----
    float* xs = &Xs[s][xr * 20 + xk];
    const float* xp = &Xmat[(size_t)(bbase + xr) * ldx + kb + xk];
    if (MODE == 0) {
      // t1 = rho*z - y computed through VGPRs (DScnt path)
      const float* yp = &ybuf[(size_t)(bbase + xr) * ldx + kb + xk];
      float4 a0 = *(const float4*)xp;
      float4 a1 = *(const float4*)(xp + 4);
      float4 y0 = *(const float4*)yp;
      float4 y1 = *(const float4*)(yp + 4);
      float4 r0 = *(const float4*)&rho[kb + xk];
      float4 r1 = *(const float4*)&rho[kb + xk + 4];
      *(float4*)xs = make_float4(r0.x * a0.x - y0.x, r0.y * a0.y - y0.y,
                                 r0.z * a0.z - y0.z, r0.w * a0.w - y0.w);
      *(float4*)(xs + 4) = make_float4(r1.x * a1.x - y1.x, r1.y * a1.y - y1.y,
                                       r1.z * a1.z - y1.z, r1.w * a1.w - y1.w);
    } else {
      async_copy_b128(xp, xs);
      async_copy_b128(xp + 4, xs + 4);
    }
    // ---- W tile (16 x 64) transposed into [n][k] via per-lane LDS scatter --
    const int gn = n0 + wn;
    const float* wp = &Wmat[(size_t)(kb + wk) * ldw + gn];
    float* wsb = &Ws[s][0];
    if (gn + 3 < N) {
      async_copy_b32(wp + 0, &wsb[(wn + 0) * 20 + wk]);
      async_copy_b32(wp + 1, &wsb[(wn + 1) * 20 + wk]);
      async_copy_b32(wp + 2, &wsb[(wn + 2) * 20 + wk]);
      async_copy_b32(wp + 3, &wsb[(wn + 3) * 20 + wk]);
    } else {
      for (int c = 0; c < 4; ++c)
        wsb[(wn + c) * 20 + wk] = (gn + c < N) ? wp[c] : 0.f;
    }
  };

  const int nk = K >> 4;
  stage(0, 0);

  const int arow = (wave << 4) + (lane & 15);
  const int kof = (lane >> 4) << 1;   // lanes 16-31 hold K+2,K+3 per ISA

  for (int i = 0; i < nk; ++i) {
    const int cur = i & 1;
    async_wait_all();        // this wave's async fills of buf[cur] are done
    __syncthreads();         // all waves filled buf[cur]; done reading other buf
    if (i + 1 < nk) stage(cur ^ 1, (i + 1) << 4);
    const float* xb = &Xs[cur][0];
    const float* wb = &Ws[cur][0];
#pragma unroll
    for (int kk = 0; kk < 16; kk += 4) {
      v2f a = *(const v2f*)&xb[arow * 20 + kk + kof];
      v2f b0 = *(const v2f*)&wb[((lane & 15) + 0) * 20 + kk + kof];
      v2f b1 = *(const v2f*)&wb[((lane & 15) + 16) * 20 + kk + kof];
      v2f b2 = *(const v2f*)&wb[((lane & 15) + 32) * 20 + kk + kof];
      v2f b3 = *(const v2f*)&wb[((lane & 15) + 48) * 20 + kk + kof];
      acc0 = __builtin_amdgcn_wmma_f32_16x16x4_f32(false, a, false, b0, (short)0, acc0, false, false);
      acc1 = __builtin_amdgcn_wmma_f32_16x16x4_f32(false, a, false, b1, (short)0, acc1, false, false);
      acc2 = __builtin_amdgcn_wmma_f32_16x16x4_f32(false, a, false, b2, (short)0, acc2, false, false);
      acc3 = __builtin_amdgcn_wmma_f32_16x16x4_f32(false, a, false, b3, (short)0, acc3, false, false);
    }
  }

  // ---- epilogue: C layout = VGPR v, lanes 0-15 -> M=v, lanes 16-31 -> M=v+8
  const int crow0 = bbase + (wave << 4) + ((lane >> 4) << 3);
  const int ccol0 = n0 + (lane & 15);
#pragma unroll
  for (int t = 0; t < 4; ++t) {
    v8f acc = (t == 0) ? acc0 : (t == 1) ? acc1 : (t == 2) ? acc2 : acc3;
    const int col = ccol0 + t * 16;
    if (col < N) {
#pragma unroll
      for (int v = 0; v < 8; ++v) {
        const int row = crow0 + v;
        const float val = acc[v];
        if (MODE == 0) {
          size_t o = (size_t)row * NVARP + col;
          Outp[o] = val + SIGMA_C * xprev[o] - qbuf[o];
        } else if (MODE == 1) {
          Outp[(size_t)row * NVARP + col] = val;
        } else {
          size_t o = (size_t)row * MCONP + col;
          float yv = ybuf[o];
          float t2 = val + yv * irho[col];
          float zv = fminf(fmaxf(t2, lbuf[o]), ubuf[o]);
          Outp[o] = zv;
          ybuf[o] = yv + rho[col] * (val - zv);
        }
      }
    }
  }
}

// ---------------------------------------------------------------------------
// Critic MLP: one thread per batch row (tiny: ~20 MFLOP total)
// ---------------------------------------------------------------------------
__global__ __launch_bounds__(256) void critic_kernel(
    const float* __restrict__ feat,
    const float* Ws, const float* bs, const float* Wst, const float* bst,
    const float* Wipr, const float* bipr, const float* Wp, const float* bp,
    const float* Ws1, const float* bs1, const float* Wst1, const float* bst1,
    const float* Wipr1, const float* bipr1, const float* Wp1, const float* bp1,
    const float* Wf1, const float* bf1, const float* Wf2, const float* bf2,
    float* __restrict__ out) {
  int b = blockIdx.x * 256 + threadIdx.x;
  if (b >= BATCH) return;
  const float* f = &feat[b * 12];
  float st0 = f[0], st1 = f[1], stor = f[2];
  float pr[9], pmax = -1e30f, pmin = 1e30f;
  for (int i = 0; i < 9; ++i) {
    pr[i] = f[3 + i];
    pmax = fmaxf(pmax, pr[i]); pmin = fminf(pmin, pr[i]);
  }
  float ipr0 = pr[0], ipr1 = pmax - pmin;
  float h[64];
  {
    float t[24];
    for (int o = 0; o < 24; ++o)
      t[o] = fmaxf(0.f, Ws[o * 2] * st0 + Ws[o * 2 + 1] * st1 + bs[o]);
    for (int o = 0; o < 24; ++o) {
      float s = bs1[o];
      for (int i = 0; i < 24; ++i) s += Ws1[o * 24 + i] * t[i];
      h[o] = fmaxf(0.f, s);
    }
  }
  {
    float t[8];
    for (int o = 0; o < 8; ++o) t[o] = fmaxf(0.f, Wst[o] * stor + bst[o]);
    for (int o = 0; o < 8; ++o) {
      float s = bst1[o];
      for (int i = 0; i < 8; ++i) s += Wst1[o * 8 + i] * t[i];
      h[24 + o] = fmaxf(0.f, s);
    }
  }
  {
    float t[8];
    for (int o = 0; o < 8; ++o)
      t[o] = fmaxf(0.f, Wipr[o * 2] * ipr0 + Wipr[o * 2 + 1] * ipr1 + bipr[o]);
    for (int o = 0; o < 8; ++o) {
      float s = bipr1[o];
      for (int i = 0; i < 8; ++i) s += Wipr1[o * 8 + i] * t[i];
      h[32 + o] = fmaxf(0.f, s);
    }
  }
  {
    float t[24];
    for (int o = 0; o < 24; ++o) {
      float s = bp[o];
      for (int i = 0; i < 9; ++i) s += Wp[o * 9 + i] * pr[i];
      t[o] = fmaxf(0.f, s);
    }
    for (int o = 0; o < 24; ++o) {
      float s = bp1[o];
      for (int i = 0; i < 24; ++i) s += Wp1[o * 24 + i] * t[i];
      h[40 + o] = fmaxf(0.f, s);
    }
  }
  float hh[64];
  for (int o = 0; o < 64; ++o) {
    float s = bf1[o];
    for (int i = 0; i < 64; ++i) s += Wf1[o * 64 + i] * h[i];
    hh[o] = fmaxf(0.f, s);
  }
  for (int o = 0; o < 64; ++o) {
    float s = bf2[o];
    for (int i = 0; i < 64; ++i) s += Wf2[o * 64 + i] * hh[i];
    out[(size_t)b * 64 + o] = fmaxf(0.f, s);
  }
}

__global__ void extract_mu(const float* __restrict__ x, float* __restrict__ out) {
  int b = blockIdx.x * 256 + threadIdx.x;
  if (b < BATCH) {
    out[b * 2 + 0] = x[(size_t)b * NVARP + 288];
    out[b * 2 + 1] = x[(size_t)b * NVARP + 289];
  }
}

// ---------------------------------------------------------------------------
extern "C" void kernel_launch(void* const* d_in, const int* in_sizes, int n_in,
                              void* d_out, int out_size, void* d_ws,
                              size_t ws_size, hipStream_t stream) {
  (void)in_sizes; (void)n_in; (void)out_size;
  if (ws_size < TOTAL_F * sizeof(float)) return;

  const float* feat  = (const float*)d_in[0];
  const float* Xlb_p = (const float*)d_in[1];
  const float* Xub_p = (const float*)d_in[2];
  const float* slb_p = (const float*)d_in[3];
  const float* sub_p = (const float*)d_in[4];
  const float* tgt_p = (const float*)d_in[5];
  const float* Az    = (const float*)d_in[6];
  const float* Au    = (const float*)d_in[7];
  const float* ZtoX  = (const float*)d_in[8];
  const float* XtoZ  = (const float*)d_in[9];

  float* ws   = (float*)d_ws;
  float* A    = ws + OFF_A;
  float* At   = ws + OFF_AT;
  float* Mt   = ws + OFF_MT;
  float* Mi   = ws + OFF_MI;
  float* rho  = ws + OFF_RHO;
  float* irho = ws + OFF_IRHO;
  float* q    = ws + OFF_Q;
  float* x    = ws + OFF_X;
  float* rhs  = ws + OFF_RHS;
  float* z    = ws + OFF_Z;
  float* y    = ws + OFF_Y;
  float* l    = ws + OFF_L;
  float* u    = ws + OFF_U;
  float* out  = (float*)d_out;

  zero_kernel<<<2048, 256, 0, stream>>>(ws, TOTAL_F);
  build_static<<<1, 256, 0, stream>>>(Az, Au, ZtoX, A, rho, irho);
  build_bounds_q<<<BATCH, 128, 0, stream>>>(feat, Xlb_p, Xub_p, slb_p, sub_p,
                                            tgt_p, XtoZ, l, u, q);
  build_mmat<<<(NVARP * NVARP + 255) / 256, 256, 0, stream>>>(A, rho, Mt);
  transposeA<<<(MCONP * NVARP + 255) / 256, 256, 0, stream>>>(A, At);
  gauss_jordan<<<1, 1024, 0, stream>>>(Mt, Mi);

  dim3 blk(256);
  dim3 g12(NVARP / 64, BATCH / 128);                 // (8, 8)
  dim3 g3((MCONP + 63) / 64, BATCH / 128);           // (13, 8)
  for (int it = 0; it < N_ITERS; ++it) {
    // rhs = (rho*z - y) @ A + sigma*x - q
    admm_gemm<0><<<g12, blk, 0, stream>>>(z, A, rhs, y, rho, irho, q, x, l, u,
                                          MCONP, NVARP, MCONP, NVARP);
    // x = rhs @ Minv
    admm_gemm<1><<<g12, blk, 0, stream>>>(rhs, Mi, x, nullptr, rho, irho, q, x,
                                          l, u, NVARP, NVARP, NVARP, NVARP);
    // Ax = x @ A^T ; z = clip(Ax + y/rho, l, u) ; y += rho*(Ax - z)
    admm_gemm<2><<<g3, blk, 0, stream>>>(x, At, z, y, rho, irho, q, x, l, u,
                                         NVARP, MCONP, NVARP, MCONP);
  }
  extract_mu<<<(BATCH + 255) / 256, 256, 0, stream>>>(x, out);
  critic_kernel<<<(BATCH + 255) / 256, 256, 0, stream>>>(
      feat, (const float*)d_in[10], (const float*)d_in[11],
      (const float*)d_in[12], (const float*)d_in[13], (const float*)d_in[14],
      (const float*)d_in[15], (const float*)d_in[16], (const float*)d_in[17],
      (const float*)d_in[18], (const float*)d_in[19], (const float*)d_in[20],
      (const float*)d_in[21], (const float*)d_in[22], (const float*)d_in[23],
      (const float*)d_in[24], (const float*)d_in[25], (const float*)d_in[26],
      (const float*)d_in[27], (const float*)d_in[28], (const float*)d_in[29],
      out + 2 * BATCH);
}